// WaveletTokenizer_23914377904172
// MI455X (gfx1250) — compile-verified
//
#include <hip/hip_runtime.h>

// ---------------------------------------------------------------------------
// EMAVQ forward for MI455X (gfx1250, wave32).
//   feats    : (4,16384,3) f32  -> N = 65536 rows, D = 3
//   codebook : (4096,3)    f32  -> V = 4096 codes
// Distance GEMM mapped onto V_WMMA_F32_16X16X4_F32:
//   A (16x4) = [-2*x0, -2*x1, -2*x2, 1.0]   (per feature row)
//   B (4x16) = [ e0,    e1,    e2,  ||e||^2] (per code column)
//   D = A*B + 0 = ||e||^2 - 2*x.e   (row-constant ||x||^2 dropped: argmin-safe)
// Codebook packed as float4{e0,e1,e2,norm2} = 64 KB, staged in LDS per block.
// ---------------------------------------------------------------------------

#define VQ_VOCAB        4096
#define VQ_NROWS        65536               // B*L = 4*16384
#define VQ_TILES        (VQ_VOCAB / 16)     // 256 column tiles of 16 codes
#define WAVES_PER_BLOCK 8
#define ROWS_PER_BLOCK  (WAVES_PER_BLOCK * 16)

typedef __attribute__((ext_vector_type(2))) float v2f;
typedef __attribute__((ext_vector_type(8))) float v8f;

// --------------------------- pack codebook ---------------------------------
__global__ __launch_bounds__(256) void vq_pack_kernel(const float* __restrict__ cb,
                                                      float4* __restrict__ cb4) {
  int v = blockIdx.x * 256 + threadIdx.x;
  if (v < VQ_VOCAB) {
    float e0 = cb[3 * v + 0];
    float e1 = cb[3 * v + 1];
    float e2 = cb[3 * v + 2];
    cb4[v] = make_float4(e0, e1, e2, e0 * e0 + e1 * e1 + e2 * e2);
  }
}

// --------------------------- main VQ kernel --------------------------------
__global__ __launch_bounds__(256) void vq_argmin_kernel(
    const float* __restrict__ feats, const float4* __restrict__ cb4,
    float* __restrict__ out_q, int* __restrict__ out_idx,
    float* __restrict__ partial) {
  // 64 KB LDS stage of the packed codebook (gfx1250 WGP has 320 KB).
  __shared__ float4 sCB[VQ_VOCAB];
  for (int i = threadIdx.x; i < VQ_VOCAB; i += 256) sCB[i] = cb4[i];
  __syncthreads();

  const int lane = threadIdx.x & 31;
  const int col  = lane & 15;   // column within 16-wide tile / row of A
  const int kh   = lane >> 4;   // which K-pair this lane holds ({0,1} vs {2,3})
  const int waveId  = blockIdx.x * WAVES_PER_BLOCK + (threadIdx.x >> 5);
  const int rowBase = waveId * 16;

  // A fragment: 16x4 f32, row m = rowBase+col, lane holds K = 2*kh + {0,1}.
  const int   m  = rowBase + col;
  const float x0 = feats[3 * m + 0];
  const float x1 = feats[3 * m + 1];
  const float x2 = feats[3 * m + 2];
  v2f A;
  A.x = kh ? (-2.0f * x2) : (-2.0f * x0);
  A.y = kh ? 1.0f         : (-2.0f * x1);   // K=3 slot multiplies B's ||e||^2

  v8f best;
  int bidx[8];
#pragma unroll
  for (int r = 0; r < 8; ++r) { best[r] = 3.4e38f; bidx[r] = 0; }

  // Inner loop: 256 tiles x (1 ds_load_b64 + 1 v_wmma + 8 cmp/select).
  const v2f* sB2 = (const v2f*)sCB;
#pragma unroll 4
  for (int t = 0; t < VQ_TILES; ++t) {
    const int code = t * 16 + col;
    const v2f B = sB2[2 * code + kh];     // {e0,e1} or {e2,||e||^2}
    v8f C = {0.f, 0.f, 0.f, 0.f, 0.f, 0.f, 0.f, 0.f};
    v8f D = __builtin_amdgcn_wmma_f32_16x16x4_f32(
        /*neg_a=*/false, A, /*neg_b=*/false, B,
        /*c_mod=*/(short)0, C, /*reuse_a=*/false, /*reuse_b=*/false);
#pragma unroll
    for (int r = 0; r < 8; ++r) {
      if (D[r] < best[r]) { best[r] = D[r]; bidx[r] = code; }
    }
  }

  // Per-row argmin now lives across a 16-lane half (D tile: VGPR r holds
  // M=r on lanes 0-15 and M=r+8 on lanes 16-31; N = lane&15). Reduce with
  // xor-shuffles 1/2/4/8 (stay inside the half on wave32).
  float lsum = 0.0f;
#pragma unroll
  for (int r = 0; r < 8; ++r) {
    float bv = best[r];
    int   bi = bidx[r];
#pragma unroll
    for (int mask = 1; mask <= 8; mask <<= 1) {
      const float ov = __shfl_xor(bv, mask, 32);
      const int   oi = __shfl_xor(bi, mask, 32);
      if (ov < bv || (ov == bv && oi < bi)) { bv = ov; bi = oi; }  // first-min tie-break
    }
    if (col == 0) {
      const int n = rowBase + r + 8 * kh;
      out_idx[n] = bi;
      const float4 e = sCB[bi];           // gather winning code from LDS
      out_q[3 * n + 0] = e.x;
      out_q[3 * n + 1] = e.y;
      out_q[3 * n + 2] = e.z;
      const float d0 = e.x - feats[3 * n + 0];
      const float d1 = e.y - feats[3 * n + 1];
      const float d2 = e.z - feats[3 * n + 2];
      lsum += d0 * d0 + d1 * d1 + d2 * d2;
    }
  }
  // Fold lane 16's partial into lane 0; one partial per wave (deterministic).
  lsum += __shfl_xor(lsum, 16, 32);
  if (lane == 0) partial[waveId] = lsum;
}

// --------------------- deterministic loss reduction ------------------------
__global__ __launch_bounds__(256) void vq_loss_kernel(const float* __restrict__ partial,
                                                      float* __restrict__ out_loss) {
  __shared__ float s[256];
  float acc = 0.0f;
  for (int i = threadIdx.x; i < (VQ_NROWS / 16); i += 256) acc += partial[i];
  s[threadIdx.x] = acc;
  __syncthreads();
  for (int off = 128; off > 0; off >>= 1) {
    if (threadIdx.x < off) s[threadIdx.x] += s[threadIdx.x + off];
    __syncthreads();
  }
  if (threadIdx.x == 0) {
    // loss = mean((q-x)^2) + 0.25*mean((x-q)^2) = 1.25 * sum / (N*3)
    *out_loss = s[0] * (1.25f / (float)(VQ_NROWS * 3));
  }
}

// ------------------------------- launcher ----------------------------------
extern "C" void kernel_launch(void* const* d_in, const int* in_sizes, int n_in,
                              void* d_out, int out_size, void* d_ws, size_t ws_size,
                              hipStream_t stream) {
  const float* feats    = (const float*)d_in[0];   // (4,16384,3) f32
  const float* codebook = (const float*)d_in[1];   // (4096,3)    f32

  float* out_q    = (float*)d_out;                         // 196608 floats
  int*   out_idx  = (int*)((float*)d_out + VQ_NROWS * 3);  // 65536 int32
  float* out_loss = (float*)d_out + VQ_NROWS * 3 + VQ_NROWS;

  float4* cb4     = (float4*)d_ws;                                   // 64 KB
  float*  partial = (float*)((char*)d_ws + VQ_VOCAB * sizeof(float4)); // 16 KB

  vq_pack_kernel<<<(VQ_VOCAB + 255) / 256, 256, 0, stream>>>(codebook, cb4);
  vq_argmin_kernel<<<VQ_NROWS / ROWS_PER_BLOCK, 256, 0, stream>>>(
      feats, cb4, out_q, out_idx, partial);
  vq_loss_kernel<<<1, 256, 0, stream>>>(partial, out_loss);
}